// ImpulseSolver_18932215840894
// MI455X (gfx1250) — compile-verified
//
#include <hip/hip_runtime.h>
#include <math.h>

// Problem constants (from reference)
#define BSZ       512
#define NO        128      // bodies
#define NCLD_IJ   64
#define NCLD_B    32
#define NCLD      96
#define NROW      192      // 96 contacts * 2 rows (normal, tangent)
#define LDA       193      // LDS row stride for A (odd -> conflict-friendly)
#define REGV      1e-4f
#define EPSV      1e-12f
#define VSTARS    12.5f    // -dist / DT / 8 with DT = 0.01

typedef __attribute__((ext_vector_type(2))) float v2f;
typedef __attribute__((ext_vector_type(8))) float v8f;

// LDS float words: A + rowVec + rhs + dvAcc + xs + vs + MinvD, then rowBody ints
#define SM_FLOATS (NROW*LDA + NROW*4 + NROW + 2*NO + 2*NO + 2*NO + NO)
#define SM_BYTES  (SM_FLOATS*sizeof(float) + NROW*2*sizeof(int))

__global__ __launch_bounds__(256)
void impulse_solver_kernel(const float* __restrict__ xv,
                           const float* __restrict__ Minv,
                           const float* __restrict__ blc,
                           const float* __restrict__ dist_ij,
                           const float* __restrict__ dist_bdry,
                           const float* __restrict__ mus,
                           const float* __restrict__ cors,
                           const int*   __restrict__ ij_ids,
                           const int*   __restrict__ b_ids,
                           float*       __restrict__ out)
{
    const int bat = blockIdx.x;
    const int tid = threadIdx.x;

    extern __shared__ char smem_raw[];
    float* A      = (float*)smem_raw;          // [NROW][LDA]
    float* rowVec = A + NROW*LDA;              // [NROW][2 slots][2 dims]
    float* rhs    = rowVec + NROW*4;           // [NROW]
    float* dvAcc  = rhs + NROW;                // [2*NO]
    float* xs     = dvAcc + 2*NO;              // [2*NO]
    float* vs     = xs + 2*NO;                 // [2*NO]
    float* MinvD  = vs + 2*NO;                 // [NO]
    int*  rowBody = (int*)(MinvD + NO);        // [NROW][2]

    // ---- Phase 1: load state into LDS ----
    {
        const float* base = xv + (size_t)bat * (4*NO);
        if (tid < 2*NO) {
            xs[tid]    = base[tid];
            vs[tid]    = base[2*NO + tid];
            dvAcc[tid] = 0.f;
        }
        if (tid < NO) MinvD[tid] = Minv[tid*NO + tid]; // diagonal of [NO,NO]
    }
    __syncthreads();

    // ---- Phase 2: contact frames -> sparse row descriptors ----
    if (tid < NCLD) {
        const int c = tid;
        float enx, eny, sgn1;
        int b0, b1;
        if (c < NCLD_IJ) {
            const int bi = ij_ids[(bat*NCLD_IJ + c)*2 + 0];
            const int bj = ij_ids[(bat*NCLD_IJ + c)*2 + 1];
            const float dx = xs[2*bj]   - xs[2*bi];
            const float dy = xs[2*bj+1] - xs[2*bi+1];
            const float r  = 1.f / sqrtf(dx*dx + dy*dy + EPSV);
            enx = dx*r; eny = dy*r;
            b0 = bi; b1 = bj; sgn1 = 1.f;
        } else {
            const int cb = c - NCLD_IJ;
            const int bo = b_ids[(bat*NCLD_B + cb)*2 + 0];
            const int bd = b_ids[(bat*NCLD_B + cb)*2 + 1];
            const float ca = blc[bd*3+0], cbv = blc[bd*3+1], cc = blc[bd*3+2];
            const float px = xs[2*bo], py = xs[2*bo+1];
            const float den = ca*ca + cbv*cbv;
            const float pcx = (cbv*cbv*px - ca*cbv*py - ca*cc) / den;
            const float pcy = (ca*ca*py  - ca*cbv*px - cbv*cc) / den;
            const float dx = pcx - px, dy = pcy - py;
            const float r  = 1.f / sqrtf(dx*dx + dy*dy + EPSV);
            enx = dx*r; eny = dy*r;
            b0 = bo; b1 = 0; sgn1 = 0.f;   // zero-vector second slot: contributes nothing
        }
        const float etx = -eny, ety = enx;
        const int r0 = 2*c, r1 = 2*c + 1;
        rowBody[2*r0+0] = b0; rowBody[2*r0+1] = b1;
        rowBody[2*r1+0] = b0; rowBody[2*r1+1] = b1;
        rowVec[4*r0+0] = -enx;      rowVec[4*r0+1] = -eny;
        rowVec[4*r0+2] = sgn1*enx;  rowVec[4*r0+3] = sgn1*eny;
        rowVec[4*r1+0] = -etx;      rowVec[4*r1+1] = -ety;
        rowVec[4*r1+2] = sgn1*etx;  rowVec[4*r1+3] = sgn1*ety;
    }
    __syncthreads();

    // ---- Phase 3a: rhs from relative velocities + stabilization ----
    if (tid < NCLD) {
        const int c = tid;
        const int r0 = 2*c, r1 = r0 + 1;
        float Jn = 0.f, Jt = 0.f;
        #pragma unroll
        for (int p = 0; p < 2; ++p) {
            const int bn = rowBody[2*r0+p];
            Jn += rowVec[4*r0+2*p+0]*vs[2*bn] + rowVec[4*r0+2*p+1]*vs[2*bn+1];
            const int bt = rowBody[2*r1+p];
            Jt += rowVec[4*r1+2*p+0]*vs[2*bt] + rowVec[4*r1+2*p+1]*vs[2*bt+1];
        }
        const float dist = (c < NCLD_IJ) ? dist_ij[bat*NCLD_IJ + c]
                                         : dist_bdry[bat*NCLD_B + (c - NCLD_IJ)];
        const float vstar = -dist * VSTARS;
        rhs[r0] = -(1.f + cors[bat*NCLD + c]) * Jn + vstar;
        rhs[r1] = -mus[bat*NCLD + c] * Jt;
    }

    // ---- Phase 3b: build A = J Minv J^T + REG*I using row sparsity ----
    for (int e = tid; e < NROW*NROW; e += 256) {
        const int ri = e / NROW;
        const int rj = e - ri*NROW;
        float acc = (ri == rj) ? REGV : 0.f;
        #pragma unroll
        for (int p = 0; p < 2; ++p) {
            const int bp = rowBody[2*ri+p];
            const float vx = rowVec[4*ri+2*p+0], vy = rowVec[4*ri+2*p+1];
            #pragma unroll
            for (int q = 0; q < 2; ++q) {
                if (bp == rowBody[2*rj+q])
                    acc += MinvD[bp] * (vx*rowVec[4*rj+2*q+0] + vy*rowVec[4*rj+2*q+1]);
            }
        }
        A[ri*LDA + rj] = acc;
    }
    __syncthreads();

    // ---- Phase 4: blocked Cholesky (lower), 16-wide panels, WMMA trailing SYRK ----
    const int wave = tid >> 5;
    const int lane = tid & 31;
    for (int k = 0; k < 12; ++k) {
        const int pbeg = 16*k, pend = 16*k + 16;
        // 4a: panel factorization (columns pbeg..pend-1), classic right-looking
        for (int jc = pbeg; jc < pend; ++jc) {
            if (tid == 0) A[jc*LDA + jc] = sqrtf(A[jc*LDA + jc]);
            __syncthreads();
            const float d = A[jc*LDA + jc];
            for (int i = jc + 1 + tid; i < NROW; i += 256) A[i*LDA + jc] /= d;
            __syncthreads();
            for (int cc = jc + 1; cc < pend; ++cc) {
                const float ljc = A[cc*LDA + jc];
                for (int i = cc + tid; i < NROW; i += 256)
                    A[i*LDA + cc] -= A[i*LDA + jc] * ljc;
            }
            __syncthreads();
        }
        // 4b: trailing update of lower-triangle tiles: A22 -= L21 * L21^T
        if (k < 11) {
            const int m = 11 - k;                  // block-rows in trailing matrix
            const int ntiles = m*(m+1)/2;          // lower triangle incl. diagonal
            const int mrow = lane & 15, half = lane >> 4;
            for (int t = wave; t < ntiles; t += 8) {   // wave-uniform -> EXEC all ones
                int ti = 0;
                while ((ti+1)*(ti+2)/2 <= t) ++ti;
                const int tj = t - ti*(ti+1)/2;
                const int rb = (k+1+ti)*16, cb = (k+1+tj)*16, kb = k*16;
                v8f acc8 = {};
                #pragma unroll
                for (int s = 0; s < 4; ++s) {
                    const int k0 = kb + 4*s + 2*half;
                    v2f av, bv;
                    av.x = A[(rb+mrow)*LDA + k0];
                    av.y = A[(rb+mrow)*LDA + k0 + 1];
                    bv.x = A[(cb+mrow)*LDA + k0];      // B = L21(tile j)^T
                    bv.y = A[(cb+mrow)*LDA + k0 + 1];
                    acc8 = __builtin_amdgcn_wmma_f32_16x16x4_f32(
                        false, av, false, bv, (short)0, acc8, false, false);
                }
                #pragma unroll
                for (int g = 0; g < 8; ++g)
                    A[(rb + g + 8*half)*LDA + cb + mrow] -= acc8[g];
            }
            __syncthreads();
        }
    }

    // ---- Phase 5: triangular solves (wave 0; DS ops in-order within a wave) ----
    if (tid < 32) {
        for (int j = 0; j < NROW; ++j) {            // L y = rhs
            const float yj = rhs[j] / A[j*LDA + j];
            for (int i = j + 1 + lane; i < NROW; i += 32)
                rhs[i] -= A[i*LDA + j] * yj;
            if (lane == 0) rhs[j] = yj;
        }
        for (int j = NROW - 1; j >= 0; --j) {       // L^T x = y
            const float xj = rhs[j] / A[j*LDA + j];
            for (int i = lane; i < j; i += 32)
                rhs[i] -= A[j*LDA + i] * xj;
            if (lane == 0) rhs[j] = xj;
        }
    }
    __syncthreads();

    // ---- Phase 6: dv = Minv J^T lam (LDS float atomics), write output ----
    if (tid < NROW) {
        const float l = rhs[tid];
        #pragma unroll
        for (int p = 0; p < 2; ++p) {
            const int bn = rowBody[2*tid+p];
            atomicAdd(&dvAcc[2*bn+0], rowVec[4*tid+2*p+0] * l);
            atomicAdd(&dvAcc[2*bn+1], rowVec[4*tid+2*p+1] * l);
        }
    }
    __syncthreads();

    if (tid < 2*NO) {
        const size_t ob = (size_t)bat * (4*NO);
        out[ob + tid]        = xs[tid];
        out[ob + 2*NO + tid] = vs[tid] + MinvD[tid >> 1] * dvAcc[tid];
    }
}

extern "C" void kernel_launch(void* const* d_in, const int* in_sizes, int n_in,
                              void* d_out, int out_size, void* d_ws, size_t ws_size,
                              hipStream_t stream) {
    (void)in_sizes; (void)n_in; (void)out_size; (void)d_ws; (void)ws_size;
    const float* xv        = (const float*)d_in[0];
    const float* Minv      = (const float*)d_in[1];
    const float* blc       = (const float*)d_in[2];
    // d_in[3] = ls (unused by the forward pass)
    const float* dist_ij   = (const float*)d_in[4];
    const float* dist_bdry = (const float*)d_in[5];
    const float* mus       = (const float*)d_in[6];
    const float* cors      = (const float*)d_in[7];
    const int*   ij_ids    = (const int*)d_in[8];
    const int*   b_ids     = (const int*)d_in[9];
    float* out = (float*)d_out;

    impulse_solver_kernel<<<dim3(BSZ), dim3(256), SM_BYTES, stream>>>(
        xv, Minv, blc, dist_ij, dist_bdry, mus, cors, ij_ids, b_ids, out);
}